// DynamicImageStaticPrimalDualNN_71631464562993
// MI455X (gfx1250) — compile-verified
//
#include <hip/hip_runtime.h>
#include <cmath>
#include <cstdint>

// ---------------------------------------------------------------------------
// Primal-dual TV (Chambolle-Pock), 48 iterations, field (2,160,160,16) fp32.
// One fused kernel per iteration (halo-recompute of q), state kept L2-resident
// (46 MB state vs 192 MB L2 -> HBM only sees the cold fill; runtime is launch
// + L2-bandwidth bound, hence 49 total launches instead of 97).
// CDNA5 paths: global_load_async_to_lds_b128 staging + s_wait_asynccnt,
// ds_load/store_b128 tile accesses, wave32-safe (no warp-width assumptions).
// No WMMA: the operator has zero matmul content (3-axis stencil + pointwise).
// ---------------------------------------------------------------------------

#define PX 160
#define PY 160
#define PT 16
#define SB (PX * PY * PT)   // per-batch scalar field elements (409600)
#define TILE 8              // 8x8 pixel tile per 256-thread block (4 lanes/pixel)
#define T_ITERS 48

#if defined(__has_builtin)
#if __has_builtin(__builtin_amdgcn_global_load_async_to_lds_b128) && \
    __has_builtin(__builtin_amdgcn_s_wait_asynccnt)
#define ASYNC_LDS 1
#endif
#endif
#ifndef ASYNC_LDS
#define ASYNC_LDS 0
#endif

__device__ __forceinline__ float4 ld4(const float* a) { return *(const float4*)a; }
__device__ __forceinline__ void   st4(float* a, float4 v) { *(float4*)a = v; }
__device__ __forceinline__ float  clip1(float v, float l) { return fminf(fmaxf(v, -l), l); }

#if ASYNC_LDS
// Exact pointee type from clang's diagnostic: vector_size(16) int, AS1 / AS3.
typedef int v4i_g __attribute__((vector_size(4 * sizeof(int))));

__device__ __forceinline__ void async_copy_b128(const float* g, float* l) {
  // Generic->AS1: global generic address equals AS1 address numerically.
  // Generic->AS3: low 32 bits of a generic LDS address are the LDS byte offset
  // (flat aperture truncation per CDNA5 ISA 10.2), so an integer round-trip is valid.
  __builtin_amdgcn_global_load_async_to_lds_b128(
      (__attribute__((address_space(1))) v4i_g*)(uintptr_t)g,
      (__attribute__((address_space(3))) v4i_g*)(uint32_t)(uintptr_t)l,
      0, 0);
}
#endif

// ---------------------------------------------------------------------------
// Init: p = x, x_ws = x, xbarA = x, qA = 0
// ---------------------------------------------------------------------------
__global__ void pd_init(const float* __restrict__ xin,
                        float* __restrict__ p,
                        float* __restrict__ xw,
                        float* __restrict__ xbA,
                        float* __restrict__ qA,
                        int n1) {
  int i = blockIdx.x * blockDim.x + threadIdx.x;
  if (i < n1) {
    float v = xin[i];
    p[i] = v; xw[i] = v; xbA[i] = v;
  }
  if (i < 3 * n1) qA[i] = 0.0f;
}

// ---------------------------------------------------------------------------
// One full primal-dual iteration.
//   phase 0: stage xbar_in over 10x10-pixel region (tile + [-1,+1] halo) to LDS
//   phase 1: q_new = clip(q + sigma*grad(xbar)) on 9x9 extended region
//            (interior 8x8 written to q_out; all of it kept in LDS)
//   phase 2: p_new, x1 = x0 - tau*(p_new + divT(q_new)), xbar_out
// ---------------------------------------------------------------------------
__launch_bounds__(256)
__global__ void pd_step(const float* __restrict__ xn,      // noisy input (fixed)
                        const float* __restrict__ lam,     // lambda_map
                        const float* __restrict__ xb_in,   // xbar (read)
                        const float* __restrict__ q_in,    // q     (read)
                        float* __restrict__ xb_out,        // xbar (write)
                        float* __restrict__ q_out,         // q    (write)
                        float* __restrict__ p,             // p (in-place)
                        const float* x0b,                  // x0 (read; may alias x1b)
                        float* x1b,                        // x1 (write; may alias x0b)
                        float sigma, float inv1ps, float tau,
                        float theta, float th1) {
  __shared__ float s_xb[100 * 16];      // 10x10 pixels x 16 t  (6400 B)
  __shared__ float s_q[3][81 * 16];     // 9x9 pixels x 16 t x 3 (15552 B)

  const int tid    = threadIdx.x;
  const int gx0    = blockIdx.x * TILE;
  const int gy0    = blockIdx.y * TILE;
  const int pp     = blockIdx.z;
  const int pbase  = pp * SB;           // scalar-field batch base
  const int qpbase = pp * 3 * SB;       // q / lambda batch base
  const int CST    = SB;                // channel stride inside q / lambda

  // ---- phase 0: stage xbar (10x10 pixels, circular wrap) ----
  for (int t = tid; t < 400; t += 256) {          // 100 pixels * 4 t-chunks
    const int px = t >> 2, ch = t & 3;
    const int i = px / 10, j = px - i * 10;       // region coords, origin (-1,-1)
    int wx = gx0 - 1 + i; if (wx < 0) wx += PX; if (wx >= PX) wx -= PX;
    int wy = gy0 - 1 + j; if (wy < 0) wy += PY; if (wy >= PY) wy -= PY;
    const int g = pbase + (wx * PY + wy) * PT + ch * 4;
    const int l = px * 16 + ch * 4;
#if ASYNC_LDS
    async_copy_b128(&xb_in[g], &s_xb[l]);
#else
    st4(&s_xb[l], ld4(&xb_in[g]));
#endif
  }
#if ASYNC_LDS
  __builtin_amdgcn_s_wait_asynccnt(0);
#endif
  __syncthreads();

  // ---- phase 1: q update over extended 9x9 region ----
  for (int task = tid; task < 324; task += 256) { // 81 pixels * 4 t-chunks
    const int px = task >> 2, ch = task & 3;
    const int ei = px / 9, ej = px - ei * 9;      // region-E coords, origin (-1,-1)
    const int lc = (ei * 10 + ej) * 16 + ch * 4;  // same origin as X region
    const float4 xc  = ld4(&s_xb[lc]);
    const float4 xxp = ld4(&s_xb[lc + 160]);      // (x+1): +10 pixels
    const float4 xyp = ld4(&s_xb[lc + 16]);       // (y+1): +1 pixel
    const float  xtn = s_xb[(ei * 10 + ej) * 16 + ((ch * 4 + 4) & 15)]; // t+1 wrap

    float4 gxv, gyv, gtv;
    gxv.x = xxp.x - xc.x; gxv.y = xxp.y - xc.y; gxv.z = xxp.z - xc.z; gxv.w = xxp.w - xc.w;
    gyv.x = xyp.x - xc.x; gyv.y = xyp.y - xc.y; gyv.z = xyp.z - xc.z; gyv.w = xyp.w - xc.w;
    gtv.x = xc.y - xc.x;  gtv.y = xc.z - xc.y;  gtv.z = xc.w - xc.z;  gtv.w = xtn - xc.w;

    int wx = gx0 - 1 + ei; if (wx < 0) wx += PX; if (wx >= PX) wx -= PX;
    int wy = gy0 - 1 + ej; if (wy < 0) wy += PY; if (wy >= PY) wy -= PY;
    const int qb = qpbase + (wx * PY + wy) * PT + ch * 4;

    float4 q0 = ld4(&q_in[qb]);
    float4 q1 = ld4(&q_in[qb + CST]);
    float4 q2 = ld4(&q_in[qb + 2 * CST]);
    const float4 l0 = ld4(&lam[qb]);
    const float4 l1 = ld4(&lam[qb + CST]);
    const float4 l2 = ld4(&lam[qb + 2 * CST]);

    q0.x = clip1(fmaf(sigma, gxv.x, q0.x), l0.x);
    q0.y = clip1(fmaf(sigma, gxv.y, q0.y), l0.y);
    q0.z = clip1(fmaf(sigma, gxv.z, q0.z), l0.z);
    q0.w = clip1(fmaf(sigma, gxv.w, q0.w), l0.w);
    q1.x = clip1(fmaf(sigma, gyv.x, q1.x), l1.x);
    q1.y = clip1(fmaf(sigma, gyv.y, q1.y), l1.y);
    q1.z = clip1(fmaf(sigma, gyv.z, q1.z), l1.z);
    q1.w = clip1(fmaf(sigma, gyv.w, q1.w), l1.w);
    q2.x = clip1(fmaf(sigma, gtv.x, q2.x), l2.x);
    q2.y = clip1(fmaf(sigma, gtv.y, q2.y), l2.y);
    q2.z = clip1(fmaf(sigma, gtv.z, q2.z), l2.z);
    q2.w = clip1(fmaf(sigma, gtv.w, q2.w), l2.w);

    const int lq = (ei * 9 + ej) * 16 + ch * 4;
    st4(&s_q[0][lq], q0);
    st4(&s_q[1][lq], q1);
    st4(&s_q[2][lq], q2);
    if (ei >= 1 && ei <= TILE && ej >= 1 && ej <= TILE) { // interior tile only
      st4(&q_out[qb],           q0);
      st4(&q_out[qb + CST],     q1);
      st4(&q_out[qb + 2 * CST], q2);
    }
  }
  __syncthreads();

  // ---- phase 2: p / x / xbar update (interior 8x8, 1 task per thread) ----
  {
    const int pix = tid >> 2, ch = tid & 3;
    const int ti = pix >> 3, tj = pix & 7;        // 0..7 within tile
    const int ei = ti + 1, ej = tj + 1;           // region-E coords of center
    const int lq = (ei * 9 + ej) * 16 + ch * 4;

    const float4 qx_c = ld4(&s_q[0][lq]);
    const float4 qx_m = ld4(&s_q[0][lq - 9 * 16]);   // (x-1)
    const float4 qy_c = ld4(&s_q[1][lq]);
    const float4 qy_m = ld4(&s_q[1][lq - 16]);       // (y-1)
    const float4 qt_c = ld4(&s_q[2][lq]);
    const float  qt_m = s_q[2][(ei * 9 + ej) * 16 + ((ch * 4 + 15) & 15)]; // t-1 wrap

    float4 dv; // adjoint: sum_d (roll(q_d,+1) - q_d)
    dv.x = (qx_m.x - qx_c.x) + (qy_m.x - qy_c.x) + (qt_m   - qt_c.x);
    dv.y = (qx_m.y - qx_c.y) + (qy_m.y - qy_c.y) + (qt_c.x - qt_c.y);
    dv.z = (qx_m.z - qx_c.z) + (qy_m.z - qy_c.z) + (qt_c.y - qt_c.z);
    dv.w = (qx_m.w - qx_c.w) + (qy_m.w - qy_c.w) + (qt_c.z - qt_c.w);

    const int gx = gx0 + ti, gy = gy0 + tj;
    const int b = pbase + (gx * PY + gy) * PT + ch * 4;

    const float4 xbc = ld4(&s_xb[((ti + 1) * 10 + (tj + 1)) * 16 + ch * 4]);
    float4 pv = ld4(&p[b]);
    const float4 xnv = ld4(&xn[b]);
    pv.x = (pv.x + sigma * (xbc.x - xnv.x)) * inv1ps;
    pv.y = (pv.y + sigma * (xbc.y - xnv.y)) * inv1ps;
    pv.z = (pv.z + sigma * (xbc.z - xnv.z)) * inv1ps;
    pv.w = (pv.w + sigma * (xbc.w - xnv.w)) * inv1ps;
    st4(&p[b], pv);

    const float4 x0v = ld4(&x0b[b]);
    float4 x1v;
    x1v.x = x0v.x - tau * (pv.x + dv.x);
    x1v.y = x0v.y - tau * (pv.y + dv.y);
    x1v.z = x0v.z - tau * (pv.z + dv.z);
    x1v.w = x0v.w - tau * (pv.w + dv.w);
    st4(&x1b[b], x1v);

    float4 xbn; // xbar = (1+theta)*x1 - theta*x0
    xbn.x = th1 * x1v.x - theta * x0v.x;
    xbn.y = th1 * x1v.y - theta * x0v.y;
    xbn.z = th1 * x1v.z - theta * x0v.z;
    xbn.w = th1 * x1v.w - theta * x0v.w;
    st4(&xb_out[b], xbn);
  }
}

// ---------------------------------------------------------------------------
extern "C" void kernel_launch(void* const* d_in, const int* in_sizes, int n_in,
                              void* d_out, int out_size, void* d_ws, size_t ws_size,
                              hipStream_t stream) {
  const float* d_x   = (const float*)d_in[0];  // (2,1,160,160,16) f32
  const float* d_lam = (const float*)d_in[1];  // (2,3,160,160,16) f32

  const int N1 = 2 * SB;                       // 819200 scalar-field elements
  float* ws   = (float*)d_ws;
  float* p    = ws;                            // N1
  float* xw   = ws + (size_t)N1;               // N1
  float* xbA  = ws + (size_t)2 * N1;           // N1
  float* xbB  = ws + (size_t)3 * N1;           // N1
  float* qA   = ws + (size_t)4 * N1;           // 3*N1
  float* qB   = ws + (size_t)7 * N1;           // 3*N1  -> 10*N1 floats = 32.8 MB total

  pd_init<<<(3 * N1 + 255) / 256, 256, 0, stream>>>(d_x, p, xw, xbA, qA, N1);

  const double s10 = 1.0 / (1.0 + exp(-10.0));
  const double L   = sqrt(13.0);
  const float sigma  = (float)(s10 / L);
  const float tau    = sigma;
  const float inv1ps = (float)(1.0 / (1.0 + s10 / L));
  const float theta  = (float)s10;
  const float th1    = (float)(1.0 + s10);

  dim3 grid(PX / TILE, PY / TILE, 2), blk(256);
  for (int i = 0; i < T_ITERS; ++i) {
    const float* xbin = (i & 1) ? xbB : xbA;
    float*       xbo  = (i & 1) ? xbA : xbB;
    const float* qin  = (i & 1) ? qB : qA;
    float*       qo   = (i & 1) ? qA : qB;
    float*       x1b  = (i == T_ITERS - 1) ? (float*)d_out : xw;
    pd_step<<<grid, blk, 0, stream>>>(d_x, d_lam, xbin, qin, xbo, qo,
                                      p, xw, x1b,
                                      sigma, inv1ps, tau, theta, th1);
  }
}